// Point_Embedding_62148176773699
// MI455X (gfx1250) — compile-verified
//
#include <hip/hip_runtime.h>
#include <math.h>

typedef __attribute__((ext_vector_type(2))) float v2f;
typedef __attribute__((ext_vector_type(8))) float v8f;

#define NPTS   4096
#define BATCH  8
#define KNN    20
#define CNT    (BATCH * NPTS * KNN)   // 655360
#define SLOPE  0.2f

__device__ __forceinline__ void lds_fence() {
    asm volatile("s_wait_dscnt 0" ::: "memory");
}

// ---------------------------------------------------------------------------
// Kernel 1: kNN. One block = 256 query points of one batch; all 4096 points
// of the batch staged in LDS (SoA, 48 KB). Register-resident sorted top-20.
// ---------------------------------------------------------------------------
__global__ __launch_bounds__(256) void knn_kernel(const float* __restrict__ a,
                                                  int* __restrict__ idxout) {
    __shared__ float sx[NPTS], sy[NPTS], sz[NPTS];
    const int tid   = threadIdx.x;
    const int b     = blockIdx.x >> 4;   // 16 blocks per batch
    const int chunk = blockIdx.x & 15;
    const float* ab = a + b * 3 * NPTS;

    for (int j = tid; j < NPTS; j += 256) {
        sx[j] = ab[j];
        sy[j] = ab[NPTS + j];
        sz[j] = ab[2 * NPTS + j];
    }
    __syncthreads();

    const int i = chunk * 256 + tid;
    const float px = sx[i], py = sy[i], pz = sz[i];

    float bd[KNN];
    int   bi[KNN];
#pragma unroll
    for (int s = 0; s < KNN; ++s) { bd[s] = 3.4e38f; bi[s] = 0; }

    for (int j = 0; j < NPTS; ++j) {
        float dx = sx[j] - px, dy = sy[j] - py, dz = sz[j] - pz;
        float d  = dx * dx + dy * dy + dz * dz;
        if (d < bd[KNN - 1]) {
            float nd = d; int ni = j;
#pragma unroll
            for (int s = 0; s < KNN; ++s) {
                if (nd < bd[s]) {
                    float td = bd[s]; int ti = bi[s];
                    bd[s] = nd; bi[s] = ni;
                    nd = td;    ni = ti;
                }
            }
        }
    }
    int* op = idxout + (b * NPTS + i) * KNN;
#pragma unroll
    for (int s = 0; s < KNN; ++s) op[s] = bi[s];
}

// ---------------------------------------------------------------------------
// Kernel 2: layer-1 BN statistics (sum / sumsq of raw h1 per channel).
// thread = (channel o = tid&63, pair stripe g = tid>>6); 640 blocks x 1024 pairs.
// ---------------------------------------------------------------------------
__global__ __launch_bounds__(256) void stats1_kernel(const float* __restrict__ a,
                                                     const int* __restrict__ idxbuf,
                                                     const float* __restrict__ W1,
                                                     float* __restrict__ sums) {
    __shared__ float w1l[64 * 8];
    const int tid = threadIdx.x;
    for (int i = tid; i < 64 * 6; i += 256) w1l[(i / 6) * 8 + (i % 6)] = W1[i];
    __syncthreads();

    const int o = tid & 63;
    const int g = tid >> 6;
    float w0 = w1l[o * 8 + 0], w1 = w1l[o * 8 + 1], w2 = w1l[o * 8 + 2];
    float w3 = w1l[o * 8 + 3], w4 = w1l[o * 8 + 4], w5 = w1l[o * 8 + 5];

    const int base = blockIdx.x * 1024;
    float sm = 0.0f, sq = 0.0f;
    for (int pp = g; pp < 1024; pp += 4) {
        const int p = base + pp;
        const int b = p / (NPTS * KNN);
        const int r = p - b * (NPTS * KNN);
        const int n = r / KNN;
        const int id = idxbuf[p];
        const float* ab = a + b * 3 * NPTS;
        float px = ab[n],  py = ab[NPTS + n],  pz = ab[2 * NPTS + n];
        float qx = ab[id], qy = ab[NPTS + id], qz = ab[2 * NPTS + id];
        float h = w0 * (qx - px) + w1 * (qy - py) + w2 * (qz - pz)
                + w3 * px + w4 * py + w5 * pz;
        sm += h;
        sq += h * h;
    }
    atomicAdd(&sums[o], sm);
    atomicAdd(&sums[64 + o], sq);
}

// ---------------------------------------------------------------------------
// Tiny finalize: fold BN (mean/var/gamma/beta) into scale s and shift t.
// ---------------------------------------------------------------------------
__global__ void finalize_bn(const float* __restrict__ sums,
                            const float* __restrict__ g,
                            const float* __restrict__ bb,
                            float* __restrict__ st) {
    const int o = threadIdx.x;        // 64 threads
    const float inv = 1.0f / (float)CNT;
    float mean = sums[o] * inv;
    float var  = sums[64 + o] * inv - mean * mean;
    float s    = g[o] * rsqrtf(var + 1e-5f);
    st[o]      = s;
    st[64 + o] = bb[o] - mean * s;
}

// ---------------------------------------------------------------------------
// Kernels 3/4: fused EdgeConv tile. One wave owns a 16-point tile of one
// batch; loops over k=0..19. Per iteration: lane computes 32 channels of
// h1n into padded LDS, then D(64x16) = W2(64x64) x h1n(64x16) via
// 16 K-steps x 4 M-tiles of V_WMMA_F32_16X16X4_F32.
//   MODE 0: accumulate layer-2 BN stats (shfl-reduce + atomics).
//   MODE 1: BN2 + LeakyReLU, running max over k in registers, store (B,64,N).
// ---------------------------------------------------------------------------
template <int MODE>
__global__ __launch_bounds__(256) void edge_mlp_kernel(
        const float* __restrict__ a, const int* __restrict__ knn,
        const float* __restrict__ W1, const float* __restrict__ W2,
        const float* __restrict__ s1t1, const float* __restrict__ s2t2,
        float* __restrict__ sums2, float* __restrict__ out) {
    __shared__ float w2l[64 * 65];        // padded stride 65 -> conflict-free
    __shared__ float w1l[64 * 8];
    __shared__ float s1l[64], t1l[64], s2l[64], t2l[64];
    __shared__ float h1buf[8 * 16 * 65];  // per-wave 16 points x 64 ch (pad 65)

    const int tid = threadIdx.x;
    for (int i = tid; i < 64 * 64; i += 256) w2l[(i >> 6) * 65 + (i & 63)] = W2[i];
    for (int i = tid; i < 64 * 6; i += 256)  w1l[(i / 6) * 8 + (i % 6)] = W1[i];
    if (tid < 64) {
        s1l[tid] = s1t1[tid];
        t1l[tid] = s1t1[64 + tid];
        if (MODE == 1) { s2l[tid] = s2t2[tid]; t2l[tid] = s2t2[64 + tid]; }
    }
    __syncthreads();

    const int wave = tid >> 5, lane = tid & 31;
    const int col  = lane & 15, half = lane >> 4;
    const int tile = blockIdx.x * 8 + wave;       // 2048 tiles total
    const int b    = tile >> 8;                   // 256 tiles per batch
    const int n0   = (tile & 255) << 4;
    const int n    = n0 + col;
    const float* ab = a + b * 3 * NPTS;
    const float px = ab[n], py = ab[NPTS + n], pz = ab[2 * NPTS + n];
    float* myh1 = h1buf + wave * (16 * 65);
    const int* myknn = knn + (b * NPTS + n) * KNN;

    float racc[4][8];
    float rsq[4][8];
#pragma unroll
    for (int m = 0; m < 4; ++m)
#pragma unroll
        for (int v = 0; v < 8; ++v) {
            racc[m][v] = (MODE == 0) ? 0.0f : -3.4e38f;
            rsq[m][v]  = 0.0f;
        }

    for (int k = 0; k < KNN; ++k) {
        const int id = myknn[k];
        const float qx = ab[id], qy = ab[NPTS + id], qz = ab[2 * NPTS + id];
        const float x0 = qx - px, x1 = qy - py, x2 = qz - pz;

        // h1n for this lane's 32 channels of point `col`
#pragma unroll
        for (int o = 0; o < 32; ++o) {
            const int oo = (half << 5) + o;
            const float* w = &w1l[oo * 8];
            float h = w[0] * x0 + w[1] * x1 + w[2] * x2
                    + w[3] * px + w[4] * py + w[5] * pz;
            h = h * s1l[oo] + t1l[oo];
            h = fmaxf(h, SLOPE * h);            // LeakyReLU (0<slope<1)
            myh1[col * 65 + oo] = h;
        }
        lds_fence();

        v8f zero = {};
        v8f acc[4];
#pragma unroll
        for (int m = 0; m < 4; ++m) acc[m] = zero;

#pragma unroll
        for (int s = 0; s < 16; ++s) {
            const int k0 = s * 4 + (half << 1);  // lanes 16-31 supply K=k0,k0+1 (4s+2,4s+3)
            v2f bm;
            bm.x = myh1[col * 65 + k0];
            bm.y = myh1[col * 65 + k0 + 1];
#pragma unroll
            for (int m = 0; m < 4; ++m) {
                const int row = m * 16 + col;
                v2f am;
                am.x = w2l[row * 65 + k0];
                am.y = w2l[row * 65 + k0 + 1];
                acc[m] = __builtin_amdgcn_wmma_f32_16x16x4_f32(
                    false, am, false, bm, (short)0, acc[m], false, false);
            }
        }
        lds_fence();

#pragma unroll
        for (int m = 0; m < 4; ++m)
#pragma unroll
            for (int v = 0; v < 8; ++v) {
                const float h2 = acc[m][v];
                if (MODE == 0) {
                    racc[m][v] += h2;
                    rsq[m][v]  += h2 * h2;
                } else {
                    const int ch = m * 16 + v + (half << 3);
                    float z = h2 * s2l[ch] + t2l[ch];
                    z = fmaxf(z, SLOPE * z);
                    racc[m][v] = fmaxf(racc[m][v], z);
                }
            }
    }

    if (MODE == 0) {
#pragma unroll
        for (int m = 0; m < 4; ++m)
#pragma unroll
            for (int v = 0; v < 8; ++v) {
                float s = racc[m][v], q = rsq[m][v];
#pragma unroll
                for (int off = 1; off < 16; off <<= 1) {
                    s += __shfl_xor(s, off, 32);
                    q += __shfl_xor(q, off, 32);
                }
                if (col == 0) {
                    const int ch = m * 16 + v + (half << 3);
                    atomicAdd(&sums2[ch], s);
                    atomicAdd(&sums2[64 + ch], q);
                }
            }
    } else {
#pragma unroll
        for (int m = 0; m < 4; ++m)
#pragma unroll
            for (int v = 0; v < 8; ++v) {
                const int ch = m * 16 + v + (half << 3);
                out[((b << 6) + ch) * NPTS + n0 + col] = racc[m][v];
            }
    }
}

// ---------------------------------------------------------------------------
extern "C" void kernel_launch(void* const* d_in, const int* in_sizes, int n_in,
                              void* d_out, int out_size, void* d_ws, size_t ws_size,
                              hipStream_t stream) {
    const float* a  = (const float*)d_in[0];
    const float* W1 = (const float*)d_in[1];
    const float* g1 = (const float*)d_in[2];
    const float* b1 = (const float*)d_in[3];
    const float* W2 = (const float*)d_in[4];
    const float* g2 = (const float*)d_in[5];
    const float* b2 = (const float*)d_in[6];
    float* out = (float*)d_out;

    char* ws = (char*)d_ws;
    int*   idxbuf = (int*)ws;                          // 655360 ints = 2.62 MB
    float* sums1  = (float*)(ws + (size_t)CNT * 4);    // 128 f
    float* s1t1   = sums1 + 128;                       // 128 f
    float* sums2  = s1t1 + 128;                        // 128 f
    float* s2t2   = sums2 + 128;                       // 128 f

    // zero the stats region (harness poisons ws with 0xAA)
    hipMemsetAsync(sums1, 0, 512 * sizeof(float), stream);

    knn_kernel<<<BATCH * (NPTS / 256), 256, 0, stream>>>(a, idxbuf);
    stats1_kernel<<<CNT / 1024, 256, 0, stream>>>(a, idxbuf, W1, sums1);
    finalize_bn<<<1, 64, 0, stream>>>(sums1, g1, b1, s1t1);
    edge_mlp_kernel<0><<<256, 256, 0, stream>>>(a, idxbuf, W1, W2, s1t1,
                                                nullptr, sums2, nullptr);
    finalize_bn<<<1, 64, 0, stream>>>(sums2, g2, b2, s2t2);
    edge_mlp_kernel<1><<<256, 256, 0, stream>>>(a, idxbuf, W1, W2, s1t1,
                                                s2t2, nullptr, out);
}